// PointwiseOrientationEncoding_57896159150407
// MI455X (gfx1250) — compile-verified
//
#include <hip/hip_runtime.h>
#include <hip/hip_bf16.h>

// Problem constants (match reference)
#define N_PTS 4096
#define B_SZ  4
#define D_CH  64
#define K_NB  32
#define P_TOT (B_SZ * N_PTS * K_NB) /* 524288 pixels */
#define R2    0.16f
#define EPS_BN 1e-5f

typedef __attribute__((ext_vector_type(2))) float v2f;
typedef __attribute__((ext_vector_type(8))) float v8f;

// ---------------------------------------------------------------- utilities
__global__ void k_zero(float* __restrict__ p, int n) {
  int i = blockIdx.x * blockDim.x + threadIdx.x;
  if (i < n) p[i] = 0.f;
}

// ---------------------------------------------------------------- ball query
// One wave32 per query point. Scan columns in chunks of 32, ballot the
// in-radius lanes, append indices in ascending order, early-exit at K.
__global__ void k_ballquery(const float* __restrict__ xyz, int* __restrict__ gidx) {
  const int tid  = threadIdx.x;
  const int lane = tid & 31;
  const int row  = blockIdx.x * 8 + (tid >> 5);   // 8 waves / block
  const int b    = row >> 12;                     // / N_PTS
  const int n    = row & (N_PTS - 1);
  const float* X = xyz + (size_t)b * 3 * N_PTS;
  const float* Y = X + N_PTS;
  const float* Z = Y + N_PTS;
  const float xn = X[n], yn = Y[n], zn = Z[n];
  int* out = gidx + (size_t)row * K_NB;

  int  cnt = 0;
  int  first = n;         // self is always in radius, so this gets overwritten
  bool haveFirst = false;
  for (int m0 = 0; m0 < N_PTS && cnt < K_NB; m0 += 32) {
    const int m = m0 + lane;
    const float dx = X[m] - xn, dy = Y[m] - yn, dz = Z[m] - zn;
    const float d2 = dx * dx + dy * dy + dz * dz;
    const bool pred = (d2 <= R2);
#if __has_builtin(__builtin_amdgcn_ballot_w32)
    unsigned mask = __builtin_amdgcn_ballot_w32(pred);
#else
    unsigned mask = (unsigned)__ballot(pred);
#endif
    if (!haveFirst && mask) { first = m0 + __builtin_ctz(mask); haveFirst = true; }
    const int pos = cnt + __popc(mask & ((1u << lane) - 1u));
    if (pred && pos < K_NB) out[pos] = m;
    cnt += __popc(mask);
  }
  // pad remaining slots with the first (smallest-index) neighbor
  for (int j = cnt + lane; j < K_NB; j += 32) out[j] = first;
}

// ---------------------------------------------------------------- layer 1
// Build feat = [rel, rel, points] (70 ch, padded to 72) for 128 pixels in LDS,
// then WMMA fp32 GEMM: x1[64][pix] = W0(64x70) * feat + b0.  Raw (pre-BN) out.
__global__ void k_conv_first(const float* __restrict__ xyz,
                             const float* __restrict__ pts,
                             const int* __restrict__ gidx,
                             const float* __restrict__ w,
                             const float* __restrict__ bias,
                             float* __restrict__ xout) {
  constexpr int CIN = 70, CINP = 72, COUT = 64;
  __shared__ float sW[COUT * CINP];
  __shared__ float sB[COUT];
  __shared__ float sF[CINP * 128];
  const int tid = threadIdx.x;

  for (int i = tid; i < COUT * CINP; i += 256) {
    const int oc = i / CINP, c = i - oc * CINP;
    sW[i] = (c < CIN) ? w[oc * CIN + c] : 0.f;
  }
  if (tid < COUT) sB[tid] = bias[tid];

  // feature construction: 2 threads per pixel (even/odd channels)
  const int base  = blockIdx.x * 128;
  const int i     = tid & 127;
  const int halfT = tid >> 7;
  const int p     = base + i;
  const int b     = p >> 17;                       // / (N*K)
  const int rem   = p & (N_PTS * K_NB - 1);
  const int n     = rem >> 5;                      // / K
  const int m     = gidx[p];
  const float* xb = xyz + (size_t)b * 3 * N_PTS;
  const float r0  = xb[m] - xb[n];
  const float r1  = xb[N_PTS + m] - xb[N_PTS + n];
  const float r2v = xb[2 * N_PTS + m] - xb[2 * N_PTS + n];
  const float* pb = pts + (size_t)b * D_CH * N_PTS;
  for (int c = halfT; c < CINP; c += 2) {
    float v;
    if (c < 6) { const int cc = (c < 3) ? c : c - 3; v = (cc == 0) ? r0 : (cc == 1) ? r1 : r2v; }
    else if (c < 70) v = pb[(size_t)(c - 6) * N_PTS + m];
    else v = 0.f;
    sF[c * 128 + i] = v;
  }
  __syncthreads();

  // WMMA: wave -> 16 pixels, loop over 4 output-channel tiles
  const int wv   = tid >> 5;
  const int lane = tid & 31;
  const int half = lane >> 4;
  const int pl   = lane & 15;
  const int px   = wv * 16 + pl;
  for (int oct = 0; oct < COUT / 16; ++oct) {
    const int ocb = oct * 16;
    v8f acc = {0.f, 0.f, 0.f, 0.f, 0.f, 0.f, 0.f, 0.f};
    for (int k0 = 0; k0 < CINP; k0 += 4) {
      const int ka = k0 + 2 * half;
      v2f a, bb;
      a.x  = sW[(ocb + pl) * CINP + ka];
      a.y  = sW[(ocb + pl) * CINP + ka + 1];
      bb.x = sF[ka * 128 + px];
      bb.y = sF[(ka + 1) * 128 + px];
      acc = __builtin_amdgcn_wmma_f32_16x16x4_f32(false, a, false, bb, (short)0, acc,
                                                  false, false);
    }
    for (int v = 0; v < 8; ++v) {
      const int oc = ocb + v + 8 * half;
      xout[(size_t)oc * P_TOT + base + px] = acc[v] + sB[oc];
    }
  }
}

// ---------------------------------------------------------------- layers 2/3
// Apply previous layer's BN (scale/shift from raw sums) + ReLU while staging
// features in LDS, then WMMA GEMM. Weights read through cache (fit in WGP$).
template <int CIN, int COUT>
__global__ void k_conv_mid(const float* __restrict__ xin,
                           const float* __restrict__ stats_in,
                           const float* __restrict__ g_in,
                           const float* __restrict__ be_in,
                           const float* __restrict__ w,
                           const float* __restrict__ bias,
                           float* __restrict__ xout) {
  __shared__ float sF[CIN * 128];
  __shared__ float sB[COUT];
  __shared__ float sA[CIN];
  __shared__ float sC[CIN];
  const int tid = threadIdx.x;
  if (tid < CIN) {
    const float s  = stats_in[2 * tid];
    const float sq = stats_in[2 * tid + 1];
    const float mu = s * (1.f / P_TOT);
    const float var = sq * (1.f / P_TOT) - mu * mu;
    const float a  = g_in[tid] * rsqrtf(var + EPS_BN);
    sA[tid] = a;
    sC[tid] = be_in[tid] - a * mu;
  }
  if (tid < COUT) sB[tid] = bias[tid];
  __syncthreads();

  const int base  = blockIdx.x * 128;
  const int i     = tid & 127;
  const int halfT = tid >> 7;
  for (int c = halfT; c < CIN; c += 2) {
    const float raw = xin[(size_t)c * P_TOT + base + i];
    sF[c * 128 + i] = fmaxf(0.f, sA[c] * raw + sC[c]);
  }
  __syncthreads();

  const int wv   = tid >> 5;
  const int lane = tid & 31;
  const int half = lane >> 4;
  const int pl   = lane & 15;
  const int px   = wv * 16 + pl;
  for (int oct = 0; oct < COUT / 16; ++oct) {
    const int ocb = oct * 16;
    v8f acc = {0.f, 0.f, 0.f, 0.f, 0.f, 0.f, 0.f, 0.f};
    for (int k0 = 0; k0 < CIN; k0 += 4) {
      const int ka = k0 + 2 * half;
      v2f a, bb;
      a.x  = w[(ocb + pl) * CIN + ka];
      a.y  = w[(ocb + pl) * CIN + ka + 1];
      bb.x = sF[ka * 128 + px];
      bb.y = sF[(ka + 1) * 128 + px];
      acc = __builtin_amdgcn_wmma_f32_16x16x4_f32(false, a, false, bb, (short)0, acc,
                                                  false, false);
    }
    for (int v = 0; v < 8; ++v) {
      const int oc = ocb + v + 8 * half;
      xout[(size_t)oc * P_TOT + base + px] = acc[v] + sB[oc];
    }
  }
}

// ---------------------------------------------------------------- BN stats
// blockIdx = c*S + s : each block reduces P/S elements of channel c.
__global__ void k_stats(const float* __restrict__ x, float* __restrict__ stats, int S) {
  const int c = blockIdx.x / S;
  const int s = blockIdx.x - c * S;
  const int chunk = P_TOT / 32;                    // S == 32
  const float* xc = x + (size_t)c * P_TOT + (size_t)s * chunk;
  float sum = 0.f, sq = 0.f;
  for (int i = threadIdx.x; i < chunk; i += 256) {
    const float v = xc[i];
    sum += v;
    sq  += v * v;
  }
  __shared__ float rs[256], rq[256];
  rs[threadIdx.x] = sum;
  rq[threadIdx.x] = sq;
  __syncthreads();
  for (int o = 128; o > 0; o >>= 1) {
    if (threadIdx.x < o) {
      rs[threadIdx.x] += rs[threadIdx.x + o];
      rq[threadIdx.x] += rq[threadIdx.x + o];
    }
    __syncthreads();
  }
  if (threadIdx.x == 0) {
    atomicAdd(&stats[2 * c], rs[0]);
    atomicAdd(&stats[2 * c + 1], rq[0]);
  }
}

// ---------------------------------------------------------------- final
// BN3 + ReLU + max over K (32 contiguous floats). out[(b*128+c)*N + n].
__global__ void k_final(const float* __restrict__ x3,
                        const float* __restrict__ stats,
                        const float* __restrict__ g,
                        const float* __restrict__ be,
                        float* __restrict__ out) {
  const int idx = blockIdx.x * 256 + threadIdx.x;  // B*128*N threads exactly
  const int n = idx & (N_PTS - 1);
  const int t = idx >> 12;
  const int c = t & 127;
  const int b = t >> 7;
  const float s  = stats[2 * c];
  const float sq = stats[2 * c + 1];
  const float mu = s * (1.f / P_TOT);
  const float var = sq * (1.f / P_TOT) - mu * mu;
  const float a  = g[c] * rsqrtf(var + EPS_BN);
  const float cc = be[c] - a * mu;
  const float* xp = x3 + (size_t)c * P_TOT + ((size_t)(b * N_PTS + n) << 5);
  float mx = 0.f;  // ReLU outputs are >= 0, so 0 is the identity for this max
  for (int k = 0; k < K_NB; ++k) mx = fmaxf(mx, a * xp[k] + cc);
  out[idx] = mx;
}

// ---------------------------------------------------------------- launcher
extern "C" void kernel_launch(void* const* d_in, const int* in_sizes, int n_in,
                              void* d_out, int out_size, void* d_ws, size_t ws_size,
                              hipStream_t stream) {
  (void)in_sizes; (void)n_in; (void)out_size; (void)ws_size;
  const float* xyz = (const float*)d_in[0];
  const float* pts = (const float*)d_in[1];
  const float* w0  = (const float*)d_in[2];
  const float* b0  = (const float*)d_in[3];
  const float* g0  = (const float*)d_in[4];
  const float* be0 = (const float*)d_in[5];
  const float* w1  = (const float*)d_in[6];
  const float* b1  = (const float*)d_in[7];
  const float* g1  = (const float*)d_in[8];
  const float* be1 = (const float*)d_in[9];
  const float* w2  = (const float*)d_in[10];
  const float* b2  = (const float*)d_in[11];
  const float* g2  = (const float*)d_in[12];
  const float* be2 = (const float*)d_in[13];
  float* out = (float*)d_out;

  char* ws = (char*)d_ws;
  int*   gidx  = (int*)ws;                                  // 8 MB
  float* stats = (float*)(ws + (8ull << 20));               // 512 floats
  float* st1 = stats;
  float* st2 = stats + 128;
  float* st3 = stats + 256;
  float* x1 = (float*)(ws + (8ull << 20) + 4096);           // 64  x P fp32
  float* x2 = x1 + (size_t)64 * P_TOT;                      // 64  x P fp32
  float* x3 = x2 + (size_t)64 * P_TOT;                      // 128 x P fp32

  k_zero<<<2, 256, 0, stream>>>(stats, 512);
  k_ballquery<<<(B_SZ * N_PTS) / 8, 256, 0, stream>>>(xyz, gidx);
  k_conv_first<<<P_TOT / 128, 256, 0, stream>>>(xyz, pts, gidx, w0, b0, x1);
  k_stats<<<64 * 32, 256, 0, stream>>>(x1, st1, 32);
  k_conv_mid<64, 64><<<P_TOT / 128, 256, 0, stream>>>(x1, st1, g0, be0, w1, b1, x2);
  k_stats<<<64 * 32, 256, 0, stream>>>(x2, st2, 32);
  k_conv_mid<64, 128><<<P_TOT / 128, 256, 0, stream>>>(x2, st2, g1, be1, w2, b2, x3);
  k_stats<<<128 * 32, 256, 0, stream>>>(x3, st3, 32);
  k_final<<<(B_SZ * 128 * N_PTS) / 256, 256, 0, stream>>>(x3, st3, g2, be2, out);
}